// TextOverRegionAttention_8349416423524
// MI455X (gfx1250) — compile-verified
//
#include <hip/hip_runtime.h>
#include <hip/hip_bf16.h>
#include <math.h>

// Problem constants (match reference)
#define B_  8
#define L_  2048
#define N_  512
#define D_  1024
#define H_  16
#define HD_ 64

typedef float v2f __attribute__((ext_vector_type(2)));
typedef float v8f __attribute__((ext_vector_type(8)));

// ---------------------------------------------------------------------------
// fp32 WMMA wrapper: D(16x16,f32) = A(16x4,f32) * B(4x16,f32) + C
// Fragment layouts (CDNA5 ISA 7.12.2):
//   A: lanes 0-15 hold rows M=0..15; VGPR v holds K = k0 + 2*(lane>=16) + v
//   B: lanes 0-15 hold cols N=0..15; VGPR v holds K = k0 + 2*(lane>=16) + v
//   C/D: VGPR r holds row M = r + 8*(lane>=16), col N = lane&15
// ---------------------------------------------------------------------------
static __device__ __forceinline__ v8f wmma_f32(v2f a, v2f b, v8f c) {
    return __builtin_amdgcn_wmma_f32_16x16x4_f32(
        /*neg_a=*/false, a, /*neg_b=*/false, b,
        /*c_mod=*/(short)0, c, /*reuse_a=*/false, /*reuse_b=*/false);
}

// ---------------------------------------------------------------------------
// LayerNorm: one 256-thread block per row of length D
// ---------------------------------------------------------------------------
__global__ __launch_bounds__(256)
void ln_kernel(const float* __restrict__ X, const float* __restrict__ g,
               const float* __restrict__ bta, float* __restrict__ Y, int D) {
    const int row = blockIdx.x;
    const float* x = X + (size_t)row * D;
    float s = 0.f, s2 = 0.f;
    for (int i = threadIdx.x; i < D; i += 256) {
        float v = x[i]; s += v; s2 += v * v;
    }
    __shared__ float red0[8], red1[8];
    const int lane = threadIdx.x & 31, wave = threadIdx.x >> 5;
    #pragma unroll
    for (int off = 16; off > 0; off >>= 1) {
        s  += __shfl_xor(s,  off, 32);
        s2 += __shfl_xor(s2, off, 32);
    }
    if (lane == 0) { red0[wave] = s; red1[wave] = s2; }
    __syncthreads();
    if (wave == 0) {
        s  = (lane < 8) ? red0[lane] : 0.f;
        s2 = (lane < 8) ? red1[lane] : 0.f;
        #pragma unroll
        for (int off = 4; off > 0; off >>= 1) {
            s  += __shfl_xor(s,  off, 32);
            s2 += __shfl_xor(s2, off, 32);
        }
        if (lane == 0) { red0[0] = s; red1[0] = s2; }
    }
    __syncthreads();
    const float mean = red0[0] / (float)D;
    const float var  = red1[0] / (float)D - mean * mean;
    const float rstd = rsqrtf(var + 1e-5f);
    float* y = Y + (size_t)row * D;
    for (int i = threadIdx.x; i < D; i += 256)
        y[i] = (x[i] - mean) * rstd * g[i] + bta[i];
}

// ---------------------------------------------------------------------------
// C[m][n] = out_scale * sum_k A[m][k] * W[n][k]  (+ bias[n] + residual[m][n])
// A: MxK row-major, W: NnxK row-major (A @ W^T), C: MxNn.
// Block = 256 threads (8 waves). Each wave owns a 32(M)x64(N) tile:
// per k-step: 2 A-frags + 4 B-frags -> 8 WMMAs (load:wmma = 0.75).
// Grid: x = M/256, y = Nn/64.
// ---------------------------------------------------------------------------
__global__ __launch_bounds__(256)
void gemm_nt_kernel(const float* __restrict__ A, const float* __restrict__ W,
                    float* __restrict__ C, int M, int K, int Nn,
                    float out_scale,
                    const float* __restrict__ bias,
                    const float* __restrict__ residual) {
    const int lane = threadIdx.x & 31;
    const int wave = threadIdx.x >> 5;
    const int hi = lane >> 4;      // 0 or 1
    const int lo = lane & 15;
    const int m0 = (blockIdx.x * 8 + wave) * 32;
    const int n0 = blockIdx.y * 64;
    if (m0 >= M) return;

    v8f acc0[4], acc1[4];
    #pragma unroll
    for (int j = 0; j < 4; ++j) { acc0[j] = {}; acc1[j] = {}; }

    const float* Arow0 = A + (size_t)(m0 + lo) * K;
    const float* Arow1 = A + (size_t)(m0 + 16 + lo) * K;
    const float* Wc0 = W + (size_t)(n0 +  0 + lo) * K;
    const float* Wc1 = W + (size_t)(n0 + 16 + lo) * K;
    const float* Wc2 = W + (size_t)(n0 + 32 + lo) * K;
    const float* Wc3 = W + (size_t)(n0 + 48 + lo) * K;

    for (int k0 = 0; k0 < K; k0 += 4) {
        const int ko = k0 + hi * 2;
        v2f a0 = *(const v2f*)(Arow0 + ko);
        v2f a1 = *(const v2f*)(Arow1 + ko);
        v2f b0 = *(const v2f*)(Wc0 + ko);
        v2f b1 = *(const v2f*)(Wc1 + ko);
        v2f b2 = *(const v2f*)(Wc2 + ko);
        v2f b3 = *(const v2f*)(Wc3 + ko);
        acc0[0] = wmma_f32(a0, b0, acc0[0]);
        acc0[1] = wmma_f32(a0, b1, acc0[1]);
        acc0[2] = wmma_f32(a0, b2, acc0[2]);
        acc0[3] = wmma_f32(a0, b3, acc0[3]);
        acc1[0] = wmma_f32(a1, b0, acc1[0]);
        acc1[1] = wmma_f32(a1, b1, acc1[1]);
        acc1[2] = wmma_f32(a1, b2, acc1[2]);
        acc1[3] = wmma_f32(a1, b3, acc1[3]);
    }

    #pragma unroll
    for (int j = 0; j < 4; ++j) {
        const int col = n0 + j * 16 + lo;
        const float bval = bias ? bias[col] : 0.f;
        #pragma unroll
        for (int r = 0; r < 8; ++r) {
            {
                const int row = m0 + r + 8 * hi;
                const size_t idx = (size_t)row * Nn + col;
                float v = acc0[j][r] * out_scale + bval;
                if (residual) v += residual[idx];
                C[idx] = v;
            }
            {
                const int row = m0 + 16 + r + 8 * hi;
                const size_t idx = (size_t)row * Nn + col;
                float v = acc1[j][r] * out_scale + bval;
                if (residual) v += residual[idx];
                C[idx] = v;
            }
        }
    }
}

// ---------------------------------------------------------------------------
// Attention: per (b, h, 32-row strip). 64 threads = 2 waves; each wave owns
// 16 query rows. Q strip held entirely in registers (16 v2f frags); S strip
// (16x512) staged in LDS, masked softmax, then P @ V.
// Q is pre-scaled by Hd^-0.5 in the Q projection.
// ---------------------------------------------------------------------------
__global__ __launch_bounds__(64)
void attn_kernel(const float* __restrict__ Q,    // (B, L, D)
                 const float* __restrict__ Km,   // (B, N, D)
                 const float* __restrict__ Vm,   // (B, N, D)
                 const unsigned char* __restrict__ mask, // (B, N) bool
                 float* __restrict__ O) {        // (B, L, D)
    __shared__ float P[2][16][N_];               // 64 KB

    const int lane = threadIdx.x & 31;
    const int wave = threadIdx.x >> 5;
    const int hi = lane >> 4;
    const int lo = lane & 15;
    const int b = blockIdx.z, h = blockIdx.y;
    const int m0 = blockIdx.x * 32 + wave * 16;

    const float* Qb = Q  + ((size_t)b * L_ + m0) * D_ + h * HD_;
    const float* Kb = Km + (size_t)b * N_ * D_ + h * HD_;
    const float* Vb = Vm + (size_t)b * N_ * D_ + h * HD_;
    const unsigned char* mb = mask + (size_t)b * N_;

    // ---- hoist Q fragments: the whole 16x64 strip = 16 v2f frags ----
    v2f qa[HD_ / 4];
    {
        const float* Arow = Qb + (size_t)lo * D_;
        #pragma unroll
        for (int k = 0; k < HD_ / 4; ++k)
            qa[k] = *(const v2f*)(Arow + k * 4 + hi * 2);
    }

    // ---- S = Q * K^T  (logits strip 16 x 512); 1 load + 1 WMMA per step ----
    for (int n0 = 0; n0 < N_; n0 += 16) {
        v8f acc = {};
        const float* Bcol = Kb + (size_t)(n0 + lo) * D_;
        #pragma unroll
        for (int k = 0; k < HD_ / 4; ++k) {
            v2f bb = *(const v2f*)(Bcol + k * 4 + hi * 2);
            acc = wmma_f32(qa[k], bb, acc);
        }
        #pragma unroll
        for (int r = 0; r < 8; ++r)
            P[wave][r + 8 * hi][n0 + lo] = acc[r];
    }

    // ---- masked softmax, 16 rows per wave, 512 cols (16 per lane) ----
    for (int r = 0; r < 16; ++r) {
        float vals[16];
        float mx = -INFINITY;
        #pragma unroll
        for (int i = 0; i < 16; ++i) {
            const int n = lane + 32 * i;
            float sv = (mb[n] != 0) ? P[wave][r][n] : -INFINITY;
            vals[i] = sv;
            mx = fmaxf(mx, sv);
        }
        #pragma unroll
        for (int off = 16; off > 0; off >>= 1)
            mx = fmaxf(mx, __shfl_xor(mx, off, 32));
        float sum = 0.f;
        #pragma unroll
        for (int i = 0; i < 16; ++i) {
            float e = (vals[i] == -INFINITY) ? 0.f : __expf(vals[i] - mx);
            vals[i] = e;
            sum += e;
        }
        #pragma unroll
        for (int off = 16; off > 0; off >>= 1)
            sum += __shfl_xor(sum, off, 32);
        const float inv = (sum > 0.f) ? (1.f / sum) : 0.f;  // all-masked -> 0
        #pragma unroll
        for (int i = 0; i < 16; ++i)
            P[wave][r][lane + 32 * i] = vals[i] * inv;
    }

    // ---- O = P * V  (16 x 64) ----
    v8f oacc[4];
    #pragma unroll
    for (int j = 0; j < 4; ++j) oacc[j] = {};
    const v2f* prow = (const v2f*)&P[wave][lo][0];
    for (int k0 = 0; k0 < N_; k0 += 4) {
        v2f a = prow[k0 / 2 + hi];
        const float* Vr0 = Vb + (size_t)(k0 + hi * 2 + 0) * D_;
        const float* Vr1 = Vb + (size_t)(k0 + hi * 2 + 1) * D_;
        #pragma unroll
        for (int j = 0; j < 4; ++j) {
            const int col = j * 16 + lo;
            v2f bb;
            bb.x = Vr0[col];
            bb.y = Vr1[col];
            oacc[j] = wmma_f32(a, bb, oacc[j]);
        }
    }
    float* Ob = O + ((size_t)b * L_ + m0) * D_ + h * HD_;
    #pragma unroll
    for (int j = 0; j < 4; ++j)
        #pragma unroll
        for (int r = 0; r < 8; ++r)
            Ob[(size_t)(r + 8 * hi) * D_ + j * 16 + lo] = oacc[j][r];
}

// ---------------------------------------------------------------------------
// Host-side orchestration
// ---------------------------------------------------------------------------
extern "C" void kernel_launch(void* const* d_in, const int* in_sizes, int n_in,
                              void* d_out, int out_size, void* d_ws, size_t ws_size,
                              hipStream_t stream) {
    (void)in_sizes; (void)n_in; (void)out_size; (void)ws_size;

    const float* text    = (const float*)d_in[0];
    const float* regions = (const float*)d_in[1];
    const unsigned char* mask = (const unsigned char*)d_in[2]; // bool array
    const float* Wq = (const float*)d_in[3];
    const float* Wk = (const float*)d_in[4];
    const float* Wv = (const float*)d_in[5];
    const float* Wo = (const float*)d_in[6];
    const float* bo = (const float*)d_in[7];
    const float* gq  = (const float*)d_in[8];
    const float* bq  = (const float*)d_in[9];
    const float* gkv = (const float*)d_in[10];
    const float* bkv = (const float*)d_in[11];
    float* out = (float*)d_out;

    const size_t TXT = (size_t)B_ * L_ * D_;   // 16.7M floats
    const size_t REG = (size_t)B_ * N_ * D_;   // 4.2M floats
    float* ws = (float*)d_ws;
    float* ln_text = ws;                        // TXT (reused later as AO)
    float* ln_reg  = ws + TXT;                  // REG (reused later as V)
    float* Qb      = ws + TXT + REG;            // TXT
    float* Kb      = ws + TXT + REG + TXT;      // REG
    float* Vb      = ln_reg;                    // alias: ln_reg dead after K GEMM
    float* AO      = ln_text;                   // alias: ln_text dead after Q GEMM

    const float qscale = 0.125f;                // Hd^-0.5 folded into Q

    // 1) pre-norms
    ln_kernel<<<B_ * L_, 256, 0, stream>>>(text,    gq,  bq,  ln_text, D_);
    ln_kernel<<<B_ * N_, 256, 0, stream>>>(regions, gkv, bkv, ln_reg,  D_);

    // 2) projections (A @ W^T via fp32 WMMA), 32x64 tile per wave
    dim3 gT(B_ * L_ / 256, D_ / 64);  // text-sized GEMM grid
    dim3 gR(B_ * N_ / 256, D_ / 64);  // region-sized GEMM grid
    gemm_nt_kernel<<<gT, 256, 0, stream>>>(ln_text, Wq, Qb, B_ * L_, D_, D_,
                                           qscale, nullptr, nullptr);
    gemm_nt_kernel<<<gR, 256, 0, stream>>>(ln_reg, Wk, Kb, B_ * N_, D_, D_,
                                           1.0f, nullptr, nullptr);
    gemm_nt_kernel<<<gR, 256, 0, stream>>>(regions, Wv, Vb, B_ * N_, D_, D_,
                                           1.0f, nullptr, nullptr);

    // 3) attention (writes AO over the dead ln_text buffer)
    dim3 gA(L_ / 32, H_, B_);
    attn_kernel<<<gA, 64, 0, stream>>>(Qb, Kb, Vb, mask, AO);

    // 4) output projection + residual + bias straight into d_out
    gemm_nt_kernel<<<gT, 256, 0, stream>>>(AO, Wo, out, B_ * L_, D_, D_,
                                           1.0f, bo, text);
}